// DenseCrossEntropyLoss_13554916786246
// MI455X (gfx1250) — compile-verified
//
#include <hip/hip_runtime.h>
#include <hip/hip_bf16.h>
#include <math.h>

typedef __attribute__((ext_vector_type(2))) float v2f;
typedef __attribute__((ext_vector_type(8))) float v8f;

#define B_    32
#define C_    256
#define HF_   64
#define WF_   64
#define M_    (HF_ * WF_)      // 4096
#define N_    64
#define INV_TEMP 100.0f        // 1 / 0.01
#define KP_STRIDE 16.0f
#define WAVES 8
#define LP    260              // padded LDS row stride (floats)
#define KSTEPS (C_ / 4)        // 64 WMMA k-steps per tile
#define KCH   8                // k-steps per pipeline chunk
#define NCH   (KSTEPS / KCH)   // 8 chunks

// ---------------------------------------------------------------------------
// Kernel 1: rnorm_trg[b*M + m] = 1 / max(||feats_trg[b,:,m]||, 1e-12)
// ---------------------------------------------------------------------------
__global__ __launch_bounds__(256) void k_rnorm_trg(const float* __restrict__ ft,
                                                   float* __restrict__ rnorm) {
  int idx = blockIdx.x * blockDim.x + threadIdx.x;
  if (idx >= B_ * M_) return;
  int b = idx / M_;
  int m = idx - b * M_;
  const float* p = ft + (size_t)b * C_ * M_ + m;
  float s = 0.0f;
#pragma unroll 8
  for (int c = 0; c < C_; ++c) {
    float v = p[(size_t)c * M_];
    s = fmaf(v, v, s);
  }
  rnorm[idx] = 1.0f / fmaxf(sqrtf(s), 1e-12f);
}

// ---------------------------------------------------------------------------
// Kernel 2: build normalized queries Q[b][n][c] and target logits l_trg[b][n].
// One block per (b, n); 256 threads = one per channel c.
// ---------------------------------------------------------------------------
__global__ __launch_bounds__(256) void k_build_q(const float* __restrict__ fs,
                                                 const float* __restrict__ ft,
                                                 const float* __restrict__ kps_src,
                                                 const float* __restrict__ kps_trg,
                                                 const float* __restrict__ rnorm_trg,
                                                 float* __restrict__ Q,
                                                 float* __restrict__ l_trg) {
  int n = blockIdx.x;
  int b = blockIdx.y;
  int c = threadIdx.x;  // C_ == 256 == blockDim.x

  float sx = kps_src[((size_t)b * N_ + n) * 2 + 0];
  float sy = kps_src[((size_t)b * N_ + n) * 2 + 1];
  float tx = kps_trg[((size_t)b * N_ + n) * 2 + 0];
  float ty = kps_trg[((size_t)b * N_ + n) * 2 + 1];
  int sfx = min(max((int)(sx / KP_STRIDE), 0), WF_ - 1);
  int sfy = min(max((int)(sy / KP_STRIDE), 0), HF_ - 1);
  int tfx = min(max((int)(tx / KP_STRIDE), 0), WF_ - 1);
  int tfy = min(max((int)(ty / KP_STRIDE), 0), HF_ - 1);
  int sidx = sfy * WF_ + sfx;
  int tidx = tfy * WF_ + tfx;

  float qv = fs[((size_t)b * C_ + c) * M_ + sidx];
  float tv = ft[((size_t)b * C_ + c) * M_ + tidx];

  __shared__ float s_sq[256];
  __shared__ float s_dt[256];
  s_sq[c] = qv * qv;
  s_dt[c] = qv * tv;
  __syncthreads();
#pragma unroll
  for (int off = 128; off > 0; off >>= 1) {
    if (c < off) {
      s_sq[c] += s_sq[c + off];
      s_dt[c] += s_dt[c + off];
    }
    __syncthreads();
  }
  float rq = 1.0f / fmaxf(sqrtf(s_sq[0]), 1e-12f);
  Q[((size_t)b * N_ + n) * C_ + c] = qv * rq;
  if (c == 0) {
    l_trg[(size_t)b * N_ + n] =
        s_dt[0] * rq * rnorm_trg[(size_t)b * M_ + tidx] * INV_TEMP;
  }
}

// ---------------------------------------------------------------------------
// Kernel 3: per (b, N-tile) block: WMMA GEMM over all M columns with fused
// online softmax (logsumexp) per row; nll = lse - l_trg.
// 8 waves; each wave covers 32 interleaved 16x16 M-tiles; K=256 in steps of 4.
// B-fragments are software-pipelined: double-buffered chunks of 8 k-steps
// (16 loads in flight) so WMMAs of chunk i overlap the loads of chunk i+1.
// ---------------------------------------------------------------------------
__global__ __launch_bounds__(256) void k_logits(const float* __restrict__ ft,
                                                const float* __restrict__ Q,
                                                const float* __restrict__ rnorm_trg,
                                                const float* __restrict__ l_trg,
                                                float* __restrict__ nll) {
  int b  = blockIdx.x;
  int nt = blockIdx.y;  // N-tile: rows nt*16 .. nt*16+15

  __shared__ float aQ[16 * LP];
  __shared__ float wmax[WAVES][16];
  __shared__ float wsum[WAVES][16];

  int tid = threadIdx.x;

  // Stage the 16 x 256 query tile into LDS (padded rows).
  const float* qsrc = Q + ((size_t)b * N_ + nt * 16) * C_;
  for (int i = tid; i < 16 * C_; i += 256) {
    int r = i >> 8;          // / C_
    int c = i & (C_ - 1);    // % C_
    aQ[r * LP + c] = qsrc[r * C_ + c];
  }
  __syncthreads();

  int wave   = tid >> 5;
  int lane   = tid & 31;
  int lane16 = lane & 15;
  int hi     = lane >> 4;    // 0: K+{0,1}, rows 0-7 of C/D ; 1: K+{2,3}, rows 8-15

  const float* Tbase = ft + (size_t)b * C_ * M_ + (size_t)(2 * hi) * M_;
  const float* Arow  = &aQ[lane16 * LP + 2 * hi];

  // A fragments: invariant across M-tiles -> keep all 64 in registers.
  v2f afrag[KSTEPS];
#pragma unroll
  for (int ks = 0; ks < KSTEPS; ++ks) {
    afrag[ks].x = Arow[4 * ks];
    afrag[ks].y = Arow[4 * ks + 1];
  }

  float rmax[8], rsum[8];
#pragma unroll
  for (int i = 0; i < 8; ++i) { rmax[i] = -__builtin_inff(); rsum[i] = 0.0f; }

  const int TILES = M_ / 16 / WAVES;  // 32
  for (int t = 0; t < TILES; ++t) {
    int m0 = (wave + t * WAVES) * 16;
    const float* Tb = Tbase + m0 + lane16;

    v8f acc = {0.f, 0.f, 0.f, 0.f, 0.f, 0.f, 0.f, 0.f};

    // ---- software-pipelined K loop: double-buffered B chunks ----
    v2f bbuf[2][KCH];
#pragma unroll
    for (int j = 0; j < KCH; ++j) {  // prologue: chunk 0
      bbuf[0][j].x = Tb[(size_t)(4 * j) * M_];
      bbuf[0][j].y = Tb[(size_t)(4 * j + 1) * M_];
    }
#pragma unroll
    for (int ch = 0; ch < NCH; ++ch) {
      const int cur = ch & 1;
      const int nxt = cur ^ 1;
      if (ch + 1 < NCH) {
#pragma unroll
        for (int j = 0; j < KCH; ++j) {
          int ks = (ch + 1) * KCH + j;
          bbuf[nxt][j].x = Tb[(size_t)(4 * ks) * M_];
          bbuf[nxt][j].y = Tb[(size_t)(4 * ks + 1) * M_];
        }
      }
#pragma unroll
      for (int j = 0; j < KCH; ++j) {
        acc = __builtin_amdgcn_wmma_f32_16x16x4_f32(
            /*neg_a=*/false, afrag[ch * KCH + j],
            /*neg_b=*/false, bbuf[cur][j],
            /*c_mod=*/(short)0, acc, /*reuse_a=*/false, /*reuse_b=*/false);
      }
    }

    float scale = rnorm_trg[(size_t)b * M_ + m0 + lane16] * INV_TEMP;
#pragma unroll
    for (int i = 0; i < 8; ++i) {
      float x  = acc[i] * scale;
      float nm = fmaxf(rmax[i], x);
      rsum[i]  = rsum[i] * __expf(rmax[i] - nm) + __expf(x - nm);
      rmax[i]  = nm;
    }
  }

  // Reduce (max,sum) across the 16 lanes of each half-wave (columns of a row).
#pragma unroll
  for (int off = 8; off >= 1; off >>= 1) {
#pragma unroll
    for (int i = 0; i < 8; ++i) {
      float om = __shfl_xor(rmax[i], off, 32);
      float os = __shfl_xor(rsum[i], off, 32);
      float nm = fmaxf(rmax[i], om);
      rsum[i]  = rsum[i] * __expf(rmax[i] - nm) + os * __expf(om - nm);
      rmax[i]  = nm;
    }
  }
  if (lane16 == 0) {
#pragma unroll
    for (int i = 0; i < 8; ++i) {
      int row = hi * 8 + i;  // C/D layout: VGPR i -> row i (lo half) / i+8 (hi half)
      wmax[wave][row] = rmax[i];
      wsum[wave][row] = rsum[i];
    }
  }
  __syncthreads();

  // Merge the 8 waves' partial logsumexp per row; emit nll.
  if (tid < 16) {
    float m = -__builtin_inff();
    float s = 0.0f;
#pragma unroll
    for (int w = 0; w < WAVES; ++w) {
      float om = wmax[w][tid];
      float os = wsum[w][tid];
      float nm = fmaxf(m, om);
      s = s * __expf(m - nm) + os * __expf(om - nm);
      m = nm;
    }
    int n   = nt * 16 + tid;
    float lse = m + __logf(s);
    nll[(size_t)b * N_ + n] = lse - l_trg[(size_t)b * N_ + n];
  }
}

// ---------------------------------------------------------------------------
// Kernel 4: masked mean over B*N -> scalar.
// ---------------------------------------------------------------------------
__global__ __launch_bounds__(256) void k_reduce(const float* __restrict__ nll,
                                                const unsigned char* __restrict__ mask,
                                                float* __restrict__ out) {
  __shared__ float s_sum[256];
  __shared__ float s_cnt[256];
  int tid = threadIdx.x;
  float s = 0.0f, c = 0.0f;
  for (int i = tid; i < B_ * N_; i += 256) {
    float m = mask[i] ? 1.0f : 0.0f;
    s += nll[i] * m;
    c += m;
  }
  s_sum[tid] = s;
  s_cnt[tid] = c;
  __syncthreads();
#pragma unroll
  for (int off = 128; off > 0; off >>= 1) {
    if (tid < off) {
      s_sum[tid] += s_sum[tid + off];
      s_cnt[tid] += s_cnt[tid + off];
    }
    __syncthreads();
  }
  if (tid == 0) out[0] = s_sum[0] / fmaxf(s_cnt[0], 1.0f);
}

// ---------------------------------------------------------------------------
extern "C" void kernel_launch(void* const* d_in, const int* in_sizes, int n_in,
                              void* d_out, int out_size, void* d_ws, size_t ws_size,
                              hipStream_t stream) {
  (void)in_sizes; (void)n_in; (void)out_size; (void)ws_size;

  const float* feats_src = (const float*)d_in[0];
  const float* feats_trg = (const float*)d_in[1];
  const float* kps_src   = (const float*)d_in[2];
  const float* kps_trg   = (const float*)d_in[3];
  const unsigned char* kps_mask = (const unsigned char*)d_in[4];
  float* out = (float*)d_out;

  // Workspace layout (floats):
  float* rnorm_trg = (float*)d_ws;                 // B*M
  float* Qbuf      = rnorm_trg + (size_t)B_ * M_;  // B*N*C
  float* ltrg      = Qbuf + (size_t)B_ * N_ * C_;  // B*N
  float* nllbuf    = ltrg + (size_t)B_ * N_;       // B*N

  // 1) target column inverse norms
  k_rnorm_trg<<<(B_ * M_ + 255) / 256, 256, 0, stream>>>(feats_trg, rnorm_trg);

  // 2) normalized queries + target logits
  k_build_q<<<dim3(N_, B_), 256, 0, stream>>>(feats_src, feats_trg, kps_src,
                                              kps_trg, rnorm_trg, Qbuf, ltrg);

  // 3) WMMA GEMM + fused online log-softmax -> per-keypoint NLL
  k_logits<<<dim3(B_, N_ / 16), 256, 0, stream>>>(feats_trg, Qbuf, rnorm_trg,
                                                  ltrg, nllbuf);

  // 4) masked mean -> scalar
  k_reduce<<<1, 256, 0, stream>>>(nllbuf, kps_mask, out);
}